// Mamba3Block_41240275976646
// MI455X (gfx1250) — compile-verified
//
#include <hip/hip_runtime.h>
#include <hip/hip_bf16.h>
#include <math.h>
#include <stdint.h>

// ---------------- problem constants (from reference) ----------------
#define D_MODEL   1280
#define D_INNER   2560
#define HEADDIM   64
#define NHEADS    40
#define D_STATE   128
#define D_CONV    4
#define CONV_DIM  2816          // D_INNER + 2*D_STATE
#define D_IN_PROJ 5416          // 2*D_INNER + 2*D_STATE + NHEADS
#define BATCH     2
#define SEQLEN    2048
#define ML        4096          // BATCH*SEQLEN

typedef float v2f __attribute__((ext_vector_type(2)));
typedef float v8f __attribute__((ext_vector_type(8)));
typedef int   v4i __attribute__((ext_vector_type(4)));

// CDNA5 async global->LDS (ASYNCcnt) path, if this toolchain exposes it.
#if defined(__has_builtin)
#if __has_builtin(__builtin_amdgcn_global_load_async_to_lds_b128)
#define HAVE_ASYNC_LDS 1
#endif
#endif
#ifndef HAVE_ASYNC_LDS
#define HAVE_ASYNC_LDS 0
#endif

#if HAVE_ASYNC_LDS
typedef __attribute__((address_space(1))) v4i as1_v4i;   // global
typedef __attribute__((address_space(3))) v4i as3_v4i;   // LDS
static __device__ __forceinline__ void async_ld_b128(const void* gsrc, void* ldst) {
    __builtin_amdgcn_global_load_async_to_lds_b128(
        (as1_v4i*)(uintptr_t)gsrc, (as3_v4i*)(uintptr_t)ldst, 0, 0);
}
static __device__ __forceinline__ void wait_async0() {
#if __has_builtin(__builtin_amdgcn_s_wait_asynccnt)
    __builtin_amdgcn_s_wait_asynccnt(0);
#else
    asm volatile("s_wait_asynccnt 0x0" ::: "memory");
#endif
}
#endif

// =====================================================================
// Kernel 1/4: fp32 GEMM on the matrix units (V_WMMA_F32_16X16X4_F32).
// C[M,N] = A[M,K] * B[K,N], row-major fp32.
// 128x64 block tile, 8 wave32 waves (4 row x 2 col), 2x2 16x16 tiles/wave.
// Double-buffered LDS staging (1 barrier/stage); A tile streamed via
// async-to-LDS (ASYNCcnt) when available, B tile via registers (stored
// transposed). Row pad = 20 floats: 16B-aligned rows (b128 LDS writes) AND
// conflict-free fragment reads (gcd(20,64)=4 -> 16 rows cover 16 banks; the
// +2 k-offset of lanes 16-31 lands in the disjoint mod-4 residue class).
// Assumes M%128==0, K%16==0 (true here); N tail guarded per block.
// =====================================================================
__global__ __launch_bounds__(256) void gemm_f32_wmma(
    float* __restrict__ C, const float* __restrict__ A,
    const float* __restrict__ B, int M, int N, int K)
{
    __shared__ float As [2][128][20];   // [buf][m][k]
    __shared__ float Bts[2][ 64][20];   // [buf][n][k] (transposed)

    const int tid  = threadIdx.x;
    const int lane = tid & 31;
    const int wave = tid >> 5;
    const int wr   = wave & 3;       // wave row 0..3 (32 rows each)
    const int wc   = wave >> 2;      // wave col 0..1 (32 cols each)
    const int hs   = lane >> 4;      // lanes 0-15 -> k0,k0+1 ; 16-31 -> k0+2,k0+3
    const int l16  = lane & 15;

    const int blockM = blockIdx.y * 128;
    const int blockN = blockIdx.x * 64;

    v8f acc[2][2] = {};

    // global->LDS staging map
    const int arow = tid >> 2;         // 0..63 (rows +0 and +64)
    const int acol = (tid & 3) * 4;    // 0,4,8,12
    const int brow = tid >> 4;         // 0..15 (k within stage)
    const int bcol = (tid & 15) * 4;   // 0..60 (n within tile)
    const bool fullN = (blockN + 64 <= N);   // uniform: tail block or not

    const int nstages = K >> 4;

    float4 bReg;
#if !HAVE_ASYNC_LDS
    float4 aReg0, aReg1;
#endif

    auto loadB = [&](int k0g) {
        const int gn = blockN + bcol;
        const float* bp = &B[(size_t)(k0g + brow) * N + gn];
        float4 r;
        if (fullN) {
            r = *(const float4*)bp;
        } else {
            r.x = (gn + 0 < N) ? bp[0] : 0.f;
            r.y = (gn + 1 < N) ? bp[1] : 0.f;
            r.z = (gn + 2 < N) ? bp[2] : 0.f;
            r.w = (gn + 3 < N) ? bp[3] : 0.f;
        }
        bReg = r;
    };
    auto storeB = [&](int buf) {
        Bts[buf][bcol + 0][brow] = bReg.x;
        Bts[buf][bcol + 1][brow] = bReg.y;
        Bts[buf][bcol + 2][brow] = bReg.z;
        Bts[buf][bcol + 3][brow] = bReg.w;
    };
#if HAVE_ASYNC_LDS
    auto loadA = [&](int k0g, int buf) {
        async_ld_b128(&A[(size_t)(blockM + arow     ) * K + k0g + acol],
                      &As[buf][arow     ][acol]);
        async_ld_b128(&A[(size_t)(blockM + arow + 64) * K + k0g + acol],
                      &As[buf][arow + 64][acol]);
    };
#else
    auto loadA = [&](int k0g, int) {
        aReg0 = *(const float4*)&A[(size_t)(blockM + arow     ) * K + k0g + acol];
        aReg1 = *(const float4*)&A[(size_t)(blockM + arow + 64) * K + k0g + acol];
    };
    auto storeA = [&](int buf) {
        As[buf][arow][acol + 0] = aReg0.x; As[buf][arow][acol + 1] = aReg0.y;
        As[buf][arow][acol + 2] = aReg0.z; As[buf][arow][acol + 3] = aReg0.w;
        As[buf][arow + 64][acol + 0] = aReg1.x; As[buf][arow + 64][acol + 1] = aReg1.y;
        As[buf][arow + 64][acol + 2] = aReg1.z; As[buf][arow + 64][acol + 3] = aReg1.w;
    };
#endif

    // ---- prologue: stage 0 into buffer 0 ----
    loadB(0);
    loadA(0, 0);
    storeB(0);
#if HAVE_ASYNC_LDS
    wait_async0();
#else
    storeA(0);
#endif
    __syncthreads();

    for (int s = 0; s < nstages; ++s) {
        const int cur = s & 1;
        const bool more = (s + 1 < nstages);
        if (more) {                 // issue next-stage loads before compute
            loadB((s + 1) << 4);
            loadA((s + 1) << 4, cur ^ 1);
        }
        // ---- 4 WMMA k-steps of depth 4 on current buffer ----
        #pragma unroll
        for (int kk = 0; kk < 4; ++kk) {
            const int kA = kk * 4 + hs * 2;
            v2f a0, a1, b0v, b1v;
            a0.x  = As [cur][wr * 32 +      l16][kA];  a0.y  = As [cur][wr * 32 +      l16][kA + 1];
            a1.x  = As [cur][wr * 32 + 16 + l16][kA];  a1.y  = As [cur][wr * 32 + 16 + l16][kA + 1];
            b0v.x = Bts[cur][wc * 32 +      l16][kA];  b0v.y = Bts[cur][wc * 32 +      l16][kA + 1];
            b1v.x = Bts[cur][wc * 32 + 16 + l16][kA];  b1v.y = Bts[cur][wc * 32 + 16 + l16][kA + 1];
            acc[0][0] = __builtin_amdgcn_wmma_f32_16x16x4_f32(false, a0, false, b0v,
                            (short)0, acc[0][0], false, false);
            acc[0][1] = __builtin_amdgcn_wmma_f32_16x16x4_f32(false, a0, false, b1v,
                            (short)0, acc[0][1], false, false);
            acc[1][0] = __builtin_amdgcn_wmma_f32_16x16x4_f32(false, a1, false, b0v,
                            (short)0, acc[1][0], false, false);
            acc[1][1] = __builtin_amdgcn_wmma_f32_16x16x4_f32(false, a1, false, b1v,
                            (short)0, acc[1][1], false, false);
        }
        if (more) {                 // commit next stage (other buffer)
            storeB(cur ^ 1);
#if HAVE_ASYNC_LDS
            wait_async0();
#else
            storeA(cur ^ 1);
#endif
        }
        __syncthreads();
    }

    // ---- store C: VGPR r holds rows (r, r+8) x col l16 per ISA layout ----
    #pragma unroll
    for (int i = 0; i < 2; ++i) {
        #pragma unroll
        for (int j = 0; j < 2; ++j) {
            #pragma unroll
            for (int r = 0; r < 8; ++r) {
                const int m = blockM + wr * 32 + i * 16 + hs * 8 + r;
                const int n = blockN + wc * 32 + j * 16 + l16;
                if (m < M && n < N) C[(size_t)m * N + n] = acc[i][j][r];
            }
        }
    }
}

// =====================================================================
// Kernel 2/4: causal depthwise conv (4-tap) + bias + SiLU on the xBC slice,
// plus dt = softplus(dt_raw + dt_bias). One block per (b,l) row.
// =====================================================================
__global__ __launch_bounds__(256) void conv_silu_dt(
    const float* __restrict__ zxbcdt, const float* __restrict__ conv_w,
    const float* __restrict__ conv_b, const float* __restrict__ dt_bias,
    float* __restrict__ xBC_act, float* __restrict__ dt_out)
{
    const int bl = blockIdx.x;          // 0..ML-1
    const int l  = bl & (SEQLEN - 1);   // position within sequence

    for (int c = threadIdx.x; c < CONV_DIM; c += 256) {
        float accv = conv_b[c];
        #pragma unroll
        for (int j = 0; j < D_CONV; ++j) {
            const int ll = l - (D_CONV - 1) + j;
            if (ll >= 0) {
                const int row = bl - (D_CONV - 1) + j;  // same batch (ll>=0)
                accv += conv_w[c * D_CONV + j] *
                        zxbcdt[(size_t)row * D_IN_PROJ + D_INNER + c];
            }
        }
        xBC_act[(size_t)bl * CONV_DIM + c] = accv / (1.f + __expf(-accv)); // SiLU
    }
    if (threadIdx.x < NHEADS) {
        const float xr = zxbcdt[(size_t)bl * D_IN_PROJ + (D_IN_PROJ - NHEADS) + threadIdx.x]
                         + dt_bias[threadIdx.x];
        dt_out[(size_t)bl * NHEADS + threadIdx.x] =
            (xr > 20.f) ? xr : log1pf(__expf(xr));      // softplus
    }
}

// =====================================================================
// Kernel 3/4: complex diagonal SSM scan, fused D-skip + SiLU(z) gate.
// One block per (batch, head). 256 threads; thread owns head element
// p = tid>>2 and state slice n in [(tid&3)*32, +32). The 4 threads sharing
// p are ADJACENT LANES of one wave -> y-reduction over D_STATE is two
// __shfl_xor steps (no LDS, no barrier). Ping-pong LDS input buffers
// remove the trailing barrier: exactly ONE barrier per timestep.
// State (64x128 complex + real bx_prev) lives in registers: 96 VGPR/thread.
// =====================================================================
__global__ __launch_bounds__(256) void mamba_scan(
    const float* __restrict__ xBC_act, const float* __restrict__ dt_ws,
    const float* __restrict__ zxbcdt, const float* __restrict__ A_log,
    const float* __restrict__ theta, const float* __restrict__ Dskip,
    float* __restrict__ yg)
{
    __shared__ float xs_s[2][HEADDIM];
    __shared__ float b_s [2][D_STATE], c_s [2][D_STATE];
    __shared__ float lre_s[2][D_STATE], lim_s[2][D_STATE];

    const int tid = threadIdx.x;
    const int b   = blockIdx.x / NHEADS;
    const int h   = blockIdx.x % NHEADS;
    const int p   = tid >> 2;            // head-dim element (0..63)
    const int grp = tid & 3;             // which 32-wide n slice
    const int nb  = grp * 32;

    const float Ah = -__expf(A_log[h]);
    const float Dh = Dskip[h];
    const float th = (tid < D_STATE) ? theta[h * D_STATE + tid] : 0.f;

    float h_re[32], h_im[32], bxp[32];
    #pragma unroll
    for (int j = 0; j < 32; ++j) { h_re[j] = 0.f; h_im[j] = 0.f; bxp[j] = 0.f; }

    for (int t = 0; t < SEQLEN; ++t) {
        const int cur = t & 1;
        const int bl  = b * SEQLEN + t;
        const float* xr = xBC_act + (size_t)bl * CONV_DIM;

        if (tid < 64)                 xs_s[cur][tid]       = xr[h * HEADDIM + tid];
        if (tid >= 64 && tid < 192)   b_s [cur][tid - 64]  = xr[D_INNER + (tid - 64)];
        if (tid >= 128)               c_s [cur][tid - 128] = xr[D_INNER + D_STATE + (tid - 128)];

        const float dtv = dt_ws[(size_t)bl * NHEADS + h];
        if (tid < D_STATE) {
            const float decay = __expf(dtv * Ah);
            const float arg   = dtv * th;
            lre_s[cur][tid] = decay * __cosf(arg);
            lim_s[cur][tid] = decay * __sinf(arg);
        }
        __syncthreads();                 // the only barrier this step

        const float xv  = xs_s[cur][p];
        const float hdt = 0.5f * dtv;
        float acc = 0.f;
        #pragma unroll
        for (int j = 0; j < 32; ++j) {
            const int n = nb + j;
            const float bx  = b_s[cur][n] * xv;
            const float lre = lre_s[cur][n], lim = lim_s[cur][n];
            const float nre = lre * h_re[j] - lim * h_im[j] + hdt * (bx + lre * bxp[j]);
            const float nim = lim * h_re[j] + lre * h_im[j] + hdt * (lim * bxp[j]);
            h_re[j] = nre; h_im[j] = nim; bxp[j] = bx;
            acc += nre * c_s[cur][n];
        }
        // reduce the 4 n-slices (adjacent lanes) without touching LDS
        acc += __shfl_xor(acc, 1, 32);
        acc += __shfl_xor(acc, 2, 32);

        if (grp == 0) {
            const float y  = acc + Dh * xv;
            const float zv = zxbcdt[(size_t)bl * D_IN_PROJ + h * HEADDIM + p];
            yg[(size_t)bl * D_INNER + h * HEADDIM + p] = y * (zv / (1.f + __expf(-zv)));
        }
        // no trailing barrier: next step writes the OTHER LDS buffer
    }
}

// =====================================================================
// kernel_launch — inputs (setup_inputs order, all fp32):
//  0 x[B,L,1280]  1 W_in[1280,5416]  2 conv_w[2816,1,4]  3 conv_b[2816]
//  4 dt_bias[40]  5 A_log[40]        6 theta[40,128]     7 Dskip[40]
//  8 W_out[2560,1280]     output: fp32 [B,L,1280]
// =====================================================================
extern "C" void kernel_launch(void* const* d_in, const int* in_sizes, int n_in,
                              void* d_out, int out_size, void* d_ws, size_t ws_size,
                              hipStream_t stream)
{
    const float* x       = (const float*)d_in[0];
    const float* W_in    = (const float*)d_in[1];
    const float* conv_w  = (const float*)d_in[2];
    const float* conv_b  = (const float*)d_in[3];
    const float* dt_bias = (const float*)d_in[4];
    const float* A_log   = (const float*)d_in[5];
    const float* theta   = (const float*)d_in[6];
    const float* Dskip   = (const float*)d_in[7];
    const float* W_out   = (const float*)d_in[8];
    float* out = (float*)d_out;

    float* ws      = (float*)d_ws;
    float* zxbcdt  = ws;                                   // ML * D_IN_PROJ
    float* xBC_act = zxbcdt  + (size_t)ML * D_IN_PROJ;     // ML * CONV_DIM
    float* dt_ws   = xBC_act + (size_t)ML * CONV_DIM;      // ML * NHEADS
    float* yg      = dt_ws   + (size_t)ML * NHEADS;        // ML * D_INNER

    // 1) in_proj GEMM: zxbcdt = x @ W_in   (4096 x 5416, K=1280)
    {
        dim3 grid((D_IN_PROJ + 63) / 64, ML / 128);
        gemm_f32_wmma<<<grid, 256, 0, stream>>>(zxbcdt, x, W_in, ML, D_IN_PROJ, D_MODEL);
    }
    // 2) depthwise conv + SiLU + softplus(dt)
    conv_silu_dt<<<ML, 256, 0, stream>>>(zxbcdt, conv_w, conv_b, dt_bias,
                                         xBC_act, dt_ws);
    // 3) sequential SSM scan, fused skip + gate
    mamba_scan<<<BATCH * NHEADS, 256, 0, stream>>>(xBC_act, dt_ws, zxbcdt,
                                                   A_log, theta, Dskip, yg);
    // 4) out_proj GEMM: out = yg @ W_out   (4096 x 1280, K=2560)
    {
        dim3 grid(D_MODEL / 64, ML / 128);
        gemm_f32_wmma<<<grid, 256, 0, stream>>>(out, yg, W_out, ML, D_MODEL, D_INNER);
    }
}